// ResidualVectorQuantizer_90031104459319
// MI455X (gfx1250) — compile-verified
//
#include <hip/hip_runtime.h>

typedef float v2f __attribute__((ext_vector_type(2)));
typedef float v8f __attribute__((ext_vector_type(8)));

#define N_ROWS   262144
#define DIMS     32
#define NQ       4
#define KCB      256
#define ROWPAD   33                      // odd pitch -> conflict-free LDS reads
#define CB_STRIDE (KCB * ROWPAD)         // 8448 floats per codebook
#define LDS_NORM_OFF (NQ * CB_STRIDE)    // 33792
#define LDS_FLOATS   (LDS_NORM_OFF + NQ * KCB) // 34816 floats = 136 KB

__global__ __launch_bounds__(256)
void rvq_kernel(const float* __restrict__ x,
                const float* __restrict__ codebooks,
                float* __restrict__ out)
{
    __shared__ float lds[LDS_FLOATS];
    const int tid = threadIdx.x;

    // ---- stage all 4 codebooks into LDS with padded rows ----
    const float4* cb4 = (const float4*)codebooks;
    for (int i4 = tid; i4 < (NQ * KCB * DIMS) / 4; i4 += 256) {
        int flat = i4 * 4;
        int q = flat >> 13;        // / (256*32)
        int k = (flat >> 5) & 255; // / 32 % 256
        int d = flat & 31;
        float4 v = cb4[i4];
        float* dst = &lds[q * CB_STRIDE + k * ROWPAD + d];
        dst[0] = v.x; dst[1] = v.y; dst[2] = v.z; dst[3] = v.w;
    }
    __syncthreads();
    // ---- precompute ||e_k||^2 ----
    for (int j = tid; j < NQ * KCB; j += 256) {
        int q = j >> 8, k = j & 255;
        const float* row = &lds[q * CB_STRIDE + k * ROWPAD];
        float s = 0.f;
        #pragma unroll
        for (int d = 0; d < DIMS; ++d) s = fmaf(row[d], row[d], s);
        lds[LDS_NORM_OFF + j] = s;
    }
    __syncthreads();

    const int wave = tid >> 5;
    const int lane = tid & 31;
    const int m    = lane & 15;          // row within 16-row tile
    const int base = (lane >> 4) << 1;   // K sub-offset: 0 (lanes 0-15) / 2 (lanes 16-31)
    const long rowTile = (long)blockIdx.x * 8 + wave;
    const long row = rowTile * 16 + m;
    const float* xrow = x + row * DIMS;

    // residual registers in A-fragment layout: r[2c],r[2c+1] = R[m][4c+base..+1]
    float r[16], xo[16];
    #pragma unroll
    for (int c = 0; c < 8; ++c) {
        float2 v = *(const float2*)(xrow + 4 * c + base);
        r[2*c] = v.x;  r[2*c+1] = v.y;
        xo[2*c] = v.x; xo[2*c+1] = v.y;
    }

    float lossAcc = 0.f;
    float* idxOut = out + (size_t)N_ROWS * DIMS + 1;

    for (int qi = 0; qi < NQ; ++qi) {
        const float* E   = &lds[qi * CB_STRIDE];
        const float* nrm = &lds[LDS_NORM_OFF + qi * KCB];

        float bestS[8]; int bestI[8];
        #pragma unroll
        for (int s = 0; s < 8; ++s) { bestS[s] = 3.4e38f; bestI[s] = 0; }

        // 16 column tiles of 16 codewords, 2 tiles per iteration:
        // two independent 8-deep WMMA accumulation chains share the A fragments.
        for (int t = 0; t < 16; t += 2) {
            const int col0 = t * 16 + m;
            const int col1 = col0 + 16;
            const float cn0 = nrm[col0];
            const float cn1 = nrm[col1];
            const float* e0 = E + col0 * ROWPAD + base;
            const float* e1 = E + col1 * ROWPAD + base;
            v8f acc0 = {};
            v8f acc1 = {};
            #pragma unroll
            for (int c = 0; c < 8; ++c) {         // K = 32 in 8 steps of 4
                v2f a;  a.x  = r[2*c];     a.y  = r[2*c+1];
                v2f b0; b0.x = e0[4*c];    b0.y = e0[4*c+1];
                v2f b1; b1.x = e1[4*c];    b1.y = e1[4*c+1];
                acc0 = __builtin_amdgcn_wmma_f32_16x16x4_f32(
                           false, a, false, b0, (short)0, acc0, false, false);
                acc1 = __builtin_amdgcn_wmma_f32_16x16x4_f32(
                           false, a, false, b1, (short)0, acc1, false, false);
            }
            // score = ||e||^2 - 2*(r.e); branchless running argmin,
            // col0 before col1 keeps first-min semantics.
            #pragma unroll
            for (int s = 0; s < 8; ++s) {
                float sc0 = fmaf(-2.f, acc0[s], cn0);
                bool t0 = sc0 < bestS[s];
                bestI[s] = t0 ? col0 : bestI[s];
                bestS[s] = t0 ? sc0  : bestS[s];
                float sc1 = fmaf(-2.f, acc1[s], cn1);
                bool t1 = sc1 < bestS[s];
                bestI[s] = t1 ? col1 : bestI[s];
                bestS[s] = t1 ? sc1  : bestS[s];
            }
        }

        // branchless min-with-index reduction inside each 16-lane half
        #pragma unroll
        for (int s = 0; s < 8; ++s) {
            #pragma unroll
            for (int off = 1; off < 16; off <<= 1) {
                float ps = __shfl_xor(bestS[s], off, 32);
                int   pi = __shfl_xor(bestI[s], off, 32);
                bool take = (ps < bestS[s]) | ((ps == bestS[s]) & (pi < bestI[s]));
                bestS[s] = take ? ps : bestS[s];
                bestI[s] = take ? pi : bestI[s];
            }
        }

        // redistribute: lane owning row m picks slot m&7 from half m>>3
        const int srcLane = (m >> 3) << 4;        // 0 or 16
        int myIdx = 0;
        #pragma unroll
        for (int s = 0; s < 8; ++s) {
            int tv = __shfl(bestI[s], srcLane, 32);
            myIdx = ((m & 7) == s) ? tv : myIdx;
        }

        if (lane < 16) idxOut[row * NQ + qi] = (float)myIdx;

        // gather chosen codeword, update residual, accumulate loss = sum(newR^2)
        const float* eq = E + myIdx * ROWPAD + base;
        #pragma unroll
        for (int c = 0; c < 8; ++c) {
            float q0 = eq[4*c], q1 = eq[4*c + 1];
            r[2*c]   -= q0;  r[2*c+1] -= q1;
            lossAcc = fmaf(r[2*c],   r[2*c],   lossAcc);
            lossAcc = fmaf(r[2*c+1], r[2*c+1], lossAcc);
        }
    }

    // x_q = x - final residual
    float* xqrow = out + row * DIMS;
    #pragma unroll
    for (int c = 0; c < 8; ++c) {
        float2 v; v.x = xo[2*c] - r[2*c]; v.y = xo[2*c+1] - r[2*c+1];
        *(float2*)(xqrow + 4 * c + base) = v;
    }

    // wave-level loss reduction; scale = (1+BETA)/(NQ*N*D)
    #pragma unroll
    for (int off = 1; off < 32; off <<= 1)
        lossAcc += __shfl_xor(lossAcc, off, 32);
    if (lane == 0)
        atomicAdd(out + (size_t)N_ROWS * DIMS, lossAcc * (1.25f / 33554432.f));
}

__global__ void rvq_zero_loss(float* out) {
    out[(size_t)N_ROWS * DIMS] = 0.f;
}

extern "C" void kernel_launch(void* const* d_in, const int* in_sizes, int n_in,
                              void* d_out, int out_size, void* d_ws, size_t ws_size,
                              hipStream_t stream) {
    const float* x  = (const float*)d_in[0];   // [262144, 32]
    const float* cb = (const float*)d_in[1];   // [4, 256, 32]
    float* out = (float*)d_out;                // x_q | mean_loss | indices

    rvq_zero_loss<<<1, 1, 0, stream>>>(out);
    // 16384 row tiles of 16 rows; 8 waves per 256-thread block -> 2048 blocks
    rvq_kernel<<<2048, 256, 0, stream>>>(x, cb, out);
}